// GraphMamba1_59863254171817
// MI455X (gfx1250) — compile-verified
//
#include <hip/hip_runtime.h>
#include <hip/hip_bf16.h>

// ---------------- problem constants (match reference) ----------------
#define Bc  4
#define Lc  2048
#define DMc 1024
#define DIc 2048      // E*DM
#define Nc  128
#define Kc  4
#define Rc  64        // DM/16
#define BLc (Bc * Lc) // 8192
#define XZW (2 * DIc) // 4096
#define DBW (Rc + 2 * Nc) // 320

typedef __attribute__((ext_vector_type(2))) float v2f;
typedef __attribute__((ext_vector_type(4))) float v4f;
typedef __attribute__((ext_vector_type(8))) float v8f;

// ---------------------------------------------------------------
// fp32 WMMA GEMM:  C[m,n] = epilogue( sum_k A[m,k] * W[n,k] )
// Each wave owns a 64(M) x 64(N) tile: 4x4 sub-tiles of 16x16,
// K stepped by 4 with V_WMMA_F32_16X16X4_F32.  Per K-step:
// 8 x b64 fragment loads -> 16 independent wmma, with the next
// step's fragments software-pipelined ahead of the compute.
// MODE bit0: scale A rows by rowscale[m] (input mask)
// MODE bit1: scale C rows by rowscale[m] (output mask)
// MODE bit2: add bias[n] then softplus (for delta)
// ---------------------------------------------------------------
template <int MODE>
__global__ __launch_bounds__(256) void mamba_gemm_wmma(
    const float* __restrict__ A, const float* __restrict__ W,
    float* __restrict__ C, int M, int N, int K,
    int lda, int ldw, int ldc,
    const float* __restrict__ rowscale, const float* __restrict__ bias)
{
    const int lane = threadIdx.x & 31;
    const int wid  = blockIdx.x * (blockDim.x >> 5) + (threadIdx.x >> 5);
    const int tiles_n = N >> 6;                 // 64-wide N tiles
    const int tm = wid / tiles_n;
    const int tn = wid - tm * tiles_n;
    const int m0 = tm << 6;                     // 64-tall M tiles
    const int n0 = tn << 6;
    if (m0 >= M) return;

    const int hg  = lane >> 4;   // half-group 0/1  (K = 2*hg + v)
    const int l16 = lane & 15;

    const float* __restrict__ ap[4];
    const float* __restrict__ wp[4];
    float ascale[4];
#pragma unroll
    for (int mi = 0; mi < 4; ++mi) {
        const int arow = m0 + (mi << 4) + l16;
        ap[mi] = A + (size_t)arow * lda + 2 * hg;
        ascale[mi] = (MODE & 1) ? rowscale[arow] : 1.0f;
    }
#pragma unroll
    for (int ti = 0; ti < 4; ++ti)
        wp[ti] = W + (size_t)(n0 + (ti << 4) + l16) * ldw + 2 * hg;

    v8f acc[4][4];
#pragma unroll
    for (int mi = 0; mi < 4; ++mi)
#pragma unroll
        for (int ti = 0; ti < 4; ++ti)
#pragma unroll
            for (int i = 0; i < 8; ++i) acc[mi][ti][i] = 0.0f;

    // software pipeline: frags for current step, next step prefetched
    v2f a[4], bf[4];
#pragma unroll
    for (int mi = 0; mi < 4; ++mi) a[mi]  = *(const v2f*)(ap[mi]);
#pragma unroll
    for (int ti = 0; ti < 4; ++ti) bf[ti] = *(const v2f*)(wp[ti]);

    for (int k0 = 0; k0 < K - 4; k0 += 4) {
        v2f a2[4], b2[4];
#pragma unroll
        for (int mi = 0; mi < 4; ++mi) a2[mi] = *(const v2f*)(ap[mi] + k0 + 4);
#pragma unroll
        for (int ti = 0; ti < 4; ++ti) b2[ti] = *(const v2f*)(wp[ti] + k0 + 4);

#pragma unroll
        for (int mi = 0; mi < 4; ++mi) {
            v2f am = a[mi];
            if (MODE & 1) am *= ascale[mi];
#pragma unroll
            for (int ti = 0; ti < 4; ++ti)
                acc[mi][ti] = __builtin_amdgcn_wmma_f32_16x16x4_f32(
                    false, am, false, bf[ti], (short)0, acc[mi][ti], false, false);
        }
#pragma unroll
        for (int mi = 0; mi < 4; ++mi) a[mi]  = a2[mi];
#pragma unroll
        for (int ti = 0; ti < 4; ++ti) bf[ti] = b2[ti];
    }
    // final K-step
#pragma unroll
    for (int mi = 0; mi < 4; ++mi) {
        v2f am = a[mi];
        if (MODE & 1) am *= ascale[mi];
#pragma unroll
        for (int ti = 0; ti < 4; ++ti)
            acc[mi][ti] = __builtin_amdgcn_wmma_f32_16x16x4_f32(
                false, am, false, bf[ti], (short)0, acc[mi][ti], false, false);
    }

    // epilogue + store
#pragma unroll
    for (int mi = 0; mi < 4; ++mi) {
#pragma unroll
        for (int ti = 0; ti < 4; ++ti) {
#pragma unroll
            for (int r = 0; r < 8; ++r) {
                const int row = m0 + (mi << 4) + r + (hg << 3);
                const int col = n0 + (ti << 4) + l16;
                float v = acc[mi][ti][r];
                if (MODE & 4) {                   // softplus(v + bias)
                    v += bias[col];
                    v = (v > 20.0f) ? v : log1pf(__expf(v));
                }
                if (MODE & 2) v *= rowscale[row];
                C[(size_t)row * ldc + col] = v;
            }
        }
    }
}

// ---------------------------------------------------------------
// attention_mask (int) -> float
// ---------------------------------------------------------------
__global__ void mamba_mask_f32(const int* __restrict__ m, float* __restrict__ mf, int n)
{
    int i = blockIdx.x * blockDim.x + threadIdx.x;
    if (i < n) mf[i] = (float)m[i];
}

// ---------------------------------------------------------------
// Depthwise causal conv (K=4) + bias + SiLU.  xi = xz[:, 0:DI].
// ---------------------------------------------------------------
__global__ __launch_bounds__(256) void mamba_conv_silu(
    const float* __restrict__ xz, const float* __restrict__ conv_w,
    const float* __restrict__ conv_b, float* __restrict__ xc)
{
    const int idx = blockIdx.x * blockDim.x + threadIdx.x;   // over BL*DI
    const int e  = idx & (DIc - 1);
    const int bl = idx >> 11;                                // /DI
    const int l  = bl & (Lc - 1);

    float acc = conv_b[e];
#pragma unroll
    for (int k = 0; k < Kc; ++k) {
        const int lk = l - (Kc - 1) + k;
        if (lk >= 0)
            acc += xz[(size_t)(bl - (Kc - 1) + k) * XZW + e] * conv_w[e * Kc + k];
    }
    xc[idx] = acc / (1.0f + __expf(-acc));                   // SiLU
}

// ---------------------------------------------------------------
// Selective scan: one wave32 per (b, d) channel; N=128 -> 4 state
// elements per lane (n = 4*lane + j so B/C are single b128 loads).
// y[b,l,d] = sum_n h[n]*C[n] via shfl_xor wave reduction.
// ---------------------------------------------------------------
__global__ __launch_bounds__(128) void mamba_scan(
    const float* __restrict__ delta, const float* __restrict__ dbc,
    const float* __restrict__ xc,    const float* __restrict__ A_log,
    float* __restrict__ y)
{
    const int lane = threadIdx.x & 31;
    const int wid  = blockIdx.x * (blockDim.x >> 5) + (threadIdx.x >> 5);
    const int b = wid / DIc;
    const int d = wid - b * DIc;

    const v4f al = *(const v4f*)(A_log + (size_t)d * Nc + 4 * lane);
    float An[4], h[4];
#pragma unroll
    for (int j = 0; j < 4; ++j) {
        An[j] = -__expf(al[j]);
        h[j] = 0.0f;
    }

    const size_t row0 = (size_t)b * Lc;
    for (int l = 0; l < Lc; ++l) {
        const size_t row = row0 + l;
        const float dt = delta[row * DIc + d];
        const float xt = xc[row * DIc + d];
        const float* __restrict__ bc = dbc + row * DBW + Rc;   // B at bc[n], C at bc[N+n]
        if (l + 1 < Lc) __builtin_prefetch(bc + DBW, 0, 1);    // next step's B/C row

        const v4f Bv = *(const v4f*)(bc + 4 * lane);
        const v4f Cv = *(const v4f*)(bc + Nc + 4 * lane);

        const float dx = dt * xt;
        float part = 0.0f;
#pragma unroll
        for (int j = 0; j < 4; ++j) {
            const float e = __expf(dt * An[j]);
            h[j] = e * h[j] + dx * Bv[j];
            part += h[j] * Cv[j];
        }
#pragma unroll
        for (int m = 16; m >= 1; m >>= 1)
            part += __shfl_xor(part, m, 32);
        if (lane == 0) y[row * DIc + d] = part;
    }
}

// ---------------------------------------------------------------
// y = (y_scan + xc*D) * silu(z),  z = xz[:, DI:2DI]
// ---------------------------------------------------------------
__global__ __launch_bounds__(256) void mamba_gate(
    const float* __restrict__ xz, const float* __restrict__ xc,
    const float* __restrict__ Dv, float* __restrict__ y)
{
    const int idx = blockIdx.x * blockDim.x + threadIdx.x;   // over BL*DI
    const int e  = idx & (DIc - 1);
    const int bl = idx >> 11;
    const float z = xz[(size_t)bl * XZW + DIc + e];
    const float v = y[idx] + xc[idx] * Dv[e];
    y[idx] = v * (z / (1.0f + __expf(-z)));
}

// ---------------------------------------------------------------
extern "C" void kernel_launch(void* const* d_in, const int* in_sizes, int n_in,
                              void* d_out, int out_size, void* d_ws, size_t ws_size,
                              hipStream_t stream)
{
    const float* x         = (const float*)d_in[0];
    const int*   attn_mask = (const int*)  d_in[1];
    const float* in_proj_w = (const float*)d_in[2];
    const float* conv_w    = (const float*)d_in[3];
    const float* conv_b    = (const float*)d_in[4];
    const float* x_proj_w  = (const float*)d_in[5];
    const float* dt_proj_w = (const float*)d_in[6];
    const float* dt_proj_b = (const float*)d_in[7];
    const float* A_log     = (const float*)d_in[8];
    const float* Dv        = (const float*)d_in[9];
    const float* out_proj_w= (const float*)d_in[10];
    float* out = (float*)d_out;

    // workspace layout (fp32)
    float* ws    = (float*)d_ws;
    float* xz    = ws;                                  // BL * 4096
    float* xc    = xz    + (size_t)BLc * XZW;           // BL * 2048
    float* dbc   = xc    + (size_t)BLc * DIc;           // BL * 320
    float* delta = dbc   + (size_t)BLc * DBW;           // BL * 2048
    float* yv    = delta + (size_t)BLc * DIc;           // BL * 2048
    float* maskf = yv    + (size_t)BLc * DIc;           // BL

    // 0) mask int -> float
    mamba_mask_f32<<<(BLc + 255) / 256, 256, 0, stream>>>(attn_mask, maskf, BLc);

    // 1) xz = (x * mask) @ in_proj_w^T       [8192 x 4096, K=1024]
    {
        const int tiles = (BLc / 64) * (XZW / 64);      // 8192 tiles
        mamba_gemm_wmma<1><<<tiles / 8, 256, 0, stream>>>(
            x, in_proj_w, xz, BLc, XZW, DMc, DMc, DMc, XZW, maskf, nullptr);
    }

    // 2) conv(K=4) + bias + silu -> xc
    mamba_conv_silu<<<(BLc * DIc) / 256, 256, 0, stream>>>(xz, conv_w, conv_b, xc);

    // 3) dbc = xc @ x_proj_w^T               [8192 x 320, K=2048]
    {
        const int tiles = (BLc / 64) * (DBW / 64);      // 640 tiles
        mamba_gemm_wmma<0><<<tiles / 8, 256, 0, stream>>>(
            xc, x_proj_w, dbc, BLc, DBW, DIc, DIc, DIc, DBW, nullptr, nullptr);
    }

    // 4) delta = softplus(dbc[:, :R] @ dt_proj_w^T + dt_proj_b)   [8192 x 2048, K=64]
    {
        const int tiles = (BLc / 64) * (DIc / 64);      // 4096 tiles
        mamba_gemm_wmma<4><<<tiles / 8, 256, 0, stream>>>(
            dbc, dt_proj_w, delta, BLc, DIc, Rc, DBW, Rc, DIc, nullptr, dt_proj_b);
    }

    // 5) selective scan -> yv  (one wave per (b,d); 8192 waves)
    mamba_scan<<<(Bc * DIc) / 4, 128, 0, stream>>>(delta, dbc, xc, A_log, yv);

    // 6) gate: yv = (yv + xc*D) * silu(z)
    mamba_gate<<<(BLc * DIc) / 256, 256, 0, stream>>>(xz, xc, Dv, yv);

    // 7) out = (yv @ out_proj_w^T) * mask    [8192 x 1024, K=2048]
    {
        const int tiles = (BLc / 64) * (DMc / 64);      // 2048 tiles
        mamba_gemm_wmma<2><<<tiles / 8, 256, 0, stream>>>(
            yv, out_proj_w, out, BLc, DMc, DIc, DIc, DIc, DMc, maskf, nullptr);
    }
}